// CategoricalLinear_2714419331414
// MI455X (gfx1250) — compile-verified
//
#include <hip/hip_runtime.h>

#define B_N   2048
#define IN_N  512
#define OUT_N 512
#define C_N   64

typedef float v2f __attribute__((ext_vector_type(2)));
typedef float v8f __attribute__((ext_vector_type(8)));

// ---- bucketing pipeline (runs in d_ws) ----------------------------------

__global__ void cl_zero_meta(int* __restrict__ counts, int* __restrict__ cursor) {
    int t = threadIdx.x;
    if (t < C_N) { counts[t] = 0; cursor[t] = 0; }
}

__global__ void cl_count(const int* __restrict__ ids, int* __restrict__ counts) {
    int b = blockIdx.x * blockDim.x + threadIdx.x;
    if (b < B_N) atomicAdd(&counts[ids[b] & (C_N - 1)], 1);
}

__global__ void cl_prefix(const int* __restrict__ counts, int* __restrict__ offsets) {
    if (threadIdx.x == 0) {
        int s = 0;
        for (int c = 0; c < C_N; ++c) { offsets[c] = s; s += counts[c]; }
    }
}

__global__ void cl_scatter(const int* __restrict__ ids, const int* __restrict__ offsets,
                           int* __restrict__ cursor, int* __restrict__ bucket) {
    int b = blockIdx.x * blockDim.x + threadIdx.x;
    if (b < B_N) {
        int c   = ids[b] & (C_N - 1);
        int pos = offsets[c] + atomicAdd(&cursor[c], 1);
        bucket[pos] = b;
    }
}

// ---- gather-GEMM per category via V_WMMA_F32_16X16X4_F32 ----------------
// Workgroup = 8 wave32s: 4 waves along N (64 cols), 2 waves along M (32 rows).
// grid = (C_N, OUT_N/64).

__global__ __launch_bounds__(256)
void cl_gemm(const float* __restrict__ x, const float* __restrict__ w,
             const int* __restrict__ counts, const int* __restrict__ offsets,
             const int* __restrict__ bucket, float* __restrict__ out) {
    const int cat  = blockIdx.x;
    const int lane = threadIdx.x & 31;
    const int wave = threadIdx.x >> 5;
    const int nSub = wave & 3;   // N sub-tile within the 64-col block
    const int mSub = wave >> 2;  // M sub-tile (0 or 1)

    const int count = counts[cat];
    const int base  = offsets[cat];

    const int lane15 = lane & 15;
    const int kq     = (lane >> 4) << 1;                 // K sub-offset: 0 or 2
    const unsigned n = blockIdx.y * 64 + nSub * 16 + lane15;

    // B-matrix base: W[cat][n][*] , contiguous along IN (K)
    const float* __restrict__ bPtr =
        w + ((size_t)cat * OUT_N + n) * (size_t)IN_N + kq;

    for (int m0 = mSub * 16; m0 < count; m0 += 32) {
        // gather one x row per lane, clamped at the ragged tail. The clamp
        // makes "invalid" tile rows duplicate row (count-1), so their WMMA
        // outputs are bitwise identical to the valid row's -- stores can be
        // unconditional (identical-value race), and EXEC stays all-ones.
        int mi   = m0 + lane15;
        int mc   = mi < count ? mi : count - 1;
        int samp = bucket[base + mc];                    // lane L: row m0+(L&15)
        const float* __restrict__ aPtr = x + (size_t)samp * IN_N + kq;
        __builtin_prefetch(aPtr, 0, 3);                  // global_prefetch_b8

        v8f acc = {};
        #pragma unroll 16
        for (int k = 0; k < IN_N; k += 4) {
            v2f a = *(const v2f*)(aPtr + k);             // A 16x4: two consecutive K
            v2f b = *(const v2f*)(bPtr + k);             // B 4x16: two consecutive K
            acc = __builtin_amdgcn_wmma_f32_16x16x4_f32(
                false, a, false, b, (short)0, acc, false, false);
        }

        // D layout: VGPR v -> tile row (8*(lane>>4) + v), N = lane&15.
        // Pull the row's (clamped) sample id from the lane that gathered it
        // via ds_bpermute; store with 32-bit offset -> saddr-form store.
        const int half8 = (lane >> 4) << 3;              // 0 or 8
        #pragma unroll
        for (int v = 0; v < 8; ++v) {
            unsigned s = (unsigned)__shfl(samp, half8 + v, 32);
            out[s * (unsigned)OUT_N + n] = acc[v];
        }
    }
}

// ---- entry ---------------------------------------------------------------

extern "C" void kernel_launch(void* const* d_in, const int* in_sizes, int n_in,
                              void* d_out, int out_size, void* d_ws, size_t ws_size,
                              hipStream_t stream) {
    const float* x   = (const float*)d_in[0];
    const int*   ids = (const int*)d_in[1];
    const float* w   = (const float*)d_in[2];
    float*       out = (float*)d_out;

    int* counts  = (int*)d_ws;
    int* offsets = counts  + C_N;
    int* cursor  = offsets + C_N;
    int* bucket  = cursor  + C_N;

    cl_zero_meta<<<1, 64, 0, stream>>>(counts, cursor);
    cl_count   <<<B_N / 256, 256, 0, stream>>>(ids, counts);
    cl_prefix  <<<1, 32, 0, stream>>>(counts, offsets);
    cl_scatter <<<B_N / 256, 256, 0, stream>>>(ids, offsets, cursor, bucket);

    dim3 grid(C_N, OUT_N / 64);
    cl_gemm<<<grid, 256, 0, stream>>>(x, w, counts, offsets, bucket, out);
}